// GC_SparseAttLayer_47347719471422
// MI455X (gfx1250) — compile-verified
//
#include <hip/hip_runtime.h>

// Problem dims (fixed by the reference)
#define BDIM 1024
#define FDIM 128
#define EDIM 128
#define KH   8
#define ODIM 128
#define LDP  132   // padded LDS row stride in floats (128 + 4 -> conflict-free WMMA frag loads)

typedef float v2f __attribute__((ext_vector_type(2)));
typedef float v8f __attribute__((ext_vector_type(8)));

// ---------------------------------------------------------------------------
// Kernel 1: kq[k,o,x] = sum_y bilinear[k,x,y] * Q[k,o,y]      (tiny: 16.8 MFLOP)
// one block per (k,o); thread x computes one output element
// ---------------------------------------------------------------------------
__global__ void kq_kernel(const float* __restrict__ Q,
                          const float* __restrict__ bil,
                          float* __restrict__ kq) {
  __shared__ float qrow[EDIM];
  const int ko = blockIdx.x;          // 0 .. K*O-1
  const int k  = ko >> 7;             // / ODIM
  const int xc = threadIdx.x;         // output x index
  qrow[xc] = Q[(size_t)ko * EDIM + xc];
  __syncthreads();
  const float* brow = bil + ((size_t)(k * EDIM + xc)) * EDIM;
  float acc = 0.f;
#pragma unroll 8
  for (int y = 0; y < EDIM; ++y) acc = fmaf(brow[y], qrow[y], acc);
  kq[(size_t)ko * EDIM + xc] = acc;
}

// ---------------------------------------------------------------------------
// Kernel 2: context[b,e] = sum_f x[b,f,e]   (coalesced column sums)
// ---------------------------------------------------------------------------
__global__ void ctx_kernel(const float* __restrict__ x, float* __restrict__ ctx) {
  const int b = blockIdx.x;
  const int e = threadIdx.x;
  const float* xb = x + (size_t)b * FDIM * EDIM + e;
  float s = 0.f;
#pragma unroll 8
  for (int f = 0; f < FDIM; ++f) s += xb[(size_t)f * EDIM];
  ctx[(size_t)b * EDIM + e] = s;
}

// ---------------------------------------------------------------------------
// Kernel 3 (fused): per (b,k) block:
//   gates(f,o) = X_b (128x128) * kq_k^T  + gc      -> v_wmma_f32_16x16x4_f32
//   entmax-bisect over f (50 iters, alpha=1.5 -> p = max(Xs-tau,0)^2)
//   out = p/sum(p) * values[k]
// 256 threads = 8 waves; wave w owns M-tile row w, all 8 N-tiles.
// ---------------------------------------------------------------------------
__global__ __launch_bounds__(256)
void fused_kernel(const float* __restrict__ x,
                  const float* __restrict__ kq,
                  const float* __restrict__ ctx,
                  const float* __restrict__ values,
                  float* __restrict__ out) {
  extern __shared__ float smem[];
  float* sX   = smem;                    // FDIM x LDP  (reused as gates[f][o] after GEMM)
  float* sKQ  = smem + FDIM * LDP;       // ODIM x LDP
  float* sCtx = smem + 2 * FDIM * LDP;   // EDIM
  float* sGc  = sCtx + EDIM;             // ODIM

  const int tid = threadIdx.x;
  const int b   = blockIdx.x >> 3;       // / KH
  const int k   = blockIdx.x & 7;

  // ---- stage x[b] (64KB) and kq[k] (64KB) into padded LDS via b128 loads ----
  const float4* gx  = (const float4*)(x  + (size_t)b * FDIM * EDIM);
  const float4* gkq = (const float4*)(kq + (size_t)k * ODIM * EDIM);
#pragma unroll
  for (int it = 0; it < 16; ++it) {
    int idx = it * 256 + tid;            // float4 index, 0..4095
    int row = idx >> 5;                  // 32 float4 per 128-float row
    int col = (idx & 31) << 2;
    float4 vx = gx[idx];
    float4 vq = gkq[idx];
    *(float4*)&sX [row * LDP + col] = vx;
    *(float4*)&sKQ[row * LDP + col] = vq;
  }
  if (tid < EDIM) sCtx[tid] = ctx[(size_t)b * EDIM + tid];
  __syncthreads();

  // ---- gc[o] = context . kq[k][o][:]  (bias for the gates) ----
  if (tid < ODIM) {
    float s = 0.f;
#pragma unroll 8
    for (int e2 = 0; e2 < EDIM; ++e2) s = fmaf(sCtx[e2], sKQ[tid * LDP + e2], s);
    sGc[tid] = s;
  }
  __syncthreads();

  // ---- GEMM: gates = X * KQ^T + gc via V_WMMA_F32_16X16X4_F32 ----
  const int wave = tid >> 5;             // M-tile row (0..7)
  const int lane = tid & 31;
  const int lm   = lane & 15;
  const int lh   = lane >> 4;

  v8f acc[8];
#pragma unroll
  for (int nt = 0; nt < 8; ++nt) {
    float g = sGc[nt * 16 + lm];         // bias depends on N (=o) only
    v8f t = {g, g, g, g, g, g, g, g};
    acc[nt] = t;
  }

  const int arow = (wave * 16 + lm) * LDP;
#pragma unroll 4
  for (int kk = 0; kk < 32; ++kk) {
    int c = kk * 4 + lh * 2;             // A: lanes0-15 K={0,1}, lanes16-31 K={2,3}
    v2f a = *(const v2f*)&sX[arow + c];
#pragma unroll
    for (int nt = 0; nt < 8; ++nt) {
      v2f bb = *(const v2f*)&sKQ[(nt * 16 + lm) * LDP + c];   // B[x][o] = KQ[o][x]
      acc[nt] = __builtin_amdgcn_wmma_f32_16x16x4_f32(
          false, a, false, bb, (short)0, acc[nt], false, false);
    }
  }

  // ---- write gates back to LDS (wave w only touches its own 16 X-rows: safe) ----
#pragma unroll
  for (int nt = 0; nt < 8; ++nt) {
    int o = nt * 16 + lm;
#pragma unroll
    for (int v = 0; v < 8; ++v) {
      int f = wave * 16 + lh * 8 + v;    // C/D layout: VGPR v -> M = v + 8*(lane>=16)
      sX[f * LDP + o] = acc[nt][v];
    }
  }
  __syncthreads();

  // ---- entmax bisection over f: thread pair (2o, 2o+1) holds one 128-row ----
  const int o    = tid >> 1;
  const int half = tid & 1;
  float xs[64];
  float mx = -3.402823466e38f;
#pragma unroll
  for (int i = 0; i < 64; ++i) {
    float g = 0.5f * sX[(half * 64 + i) * LDP + o];   // Xs = gates * (alpha-1)
    xs[i] = g;
    mx = fmaxf(mx, g);
  }
  mx = fmaxf(mx, __shfl_xor(mx, 1, 32));

  float tau_lo = mx - 1.0f;
  float tau_hi = mx - 0.08838834764831845f;           // (1/128)^0.5
  float dm     = tau_hi - tau_lo;
  float tau_m  = tau_lo;
  float ssum   = 1.0f;
  for (int it = 0; it < 50; ++it) {
    dm *= 0.5f;
    tau_m = tau_lo + dm;
    float s = 0.f;
#pragma unroll
    for (int i = 0; i < 64; ++i) {
      float p = fmaxf(xs[i] - tau_m, 0.f);
      s = fmaf(p, p, s);                               // p^(1/(alpha-1)) = p^2
    }
    ssum = s + __shfl_xor(s, 1, 32);
    if (ssum >= 1.0f) tau_lo = tau_m;                  // f = sum - 1 >= 0
  }

  // ---- normalized probabilities back to LDS (each (f,o) owned by one thread) ----
  const float inv = 1.0f / ssum;
#pragma unroll
  for (int i = 0; i < 64; ++i) {
    float p = fmaxf(xs[i] - tau_m, 0.f);
    sX[(half * 64 + i) * LDP + o] = p * p * inv;
  }
  __syncthreads();

  // ---- fused scale by values[k] + coalesced 536MB store ----
  const float* vals = values + (size_t)k * ODIM * FDIM;
  float*       outb = out + ((size_t)b * KH + k) * ODIM * FDIM;
#pragma unroll 4
  for (int rep = 0; rep < 64; ++rep) {
    int idx = rep * 256 + tid;           // = oo*FDIM + ff, ff contiguous across lanes
    int oo = idx >> 7;
    int ff = idx & 127;
    outb[idx] = sX[ff * LDP + oo] * vals[idx];
  }
}

// ---------------------------------------------------------------------------
extern "C" void kernel_launch(void* const* d_in, const int* in_sizes, int n_in,
                              void* d_out, int out_size, void* d_ws, size_t ws_size,
                              hipStream_t stream) {
  const float* x      = (const float*)d_in[0];  // (B,F,E)
  const float* Q      = (const float*)d_in[1];  // (K,O,E)
  const float* bil    = (const float*)d_in[2];  // (K,E,E)
  const float* values = (const float*)d_in[3];  // (K,O,F)
  float* out = (float*)d_out;                   // (B,K,O,F)

  float* kq  = (float*)d_ws;                    // K*O*E floats = 512 KB
  float* ctx = kq + (size_t)KH * ODIM * EDIM;   // B*E floats   = 512 KB

  kq_kernel <<<KH * ODIM, 128, 0, stream>>>(Q, bil, kq);
  ctx_kernel<<<BDIM,      128, 0, stream>>>(x, ctx);

  size_t lds = (size_t)(2 * FDIM * LDP + EDIM + ODIM) * sizeof(float);  // ~136 KB
  fused_kernel<<<BDIM * KH, 256, lds, stream>>>(x, kq, ctx, values, out);
}